// MFMCGCNABSA_89283780149584
// MI455X (gfx1250) — compile-verified
//
#include <hip/hip_runtime.h>
#include <math.h>

// Problem constants from the reference
#define NB     64
#define NS     256
#define NIN    512
#define NMEM   256
#define NHEADS 8
#define NTOPK  200

typedef __attribute__((ext_vector_type(2))) float v2f;
typedef __attribute__((ext_vector_type(8))) float v8f;
typedef int v4i __attribute__((vector_size(16)));   // matches async-builtin param type

// CDNA5 fp32 WMMA: D(16x16) = A(16x4) * B(4x16) + C. 8-arg VOP3P form.
__device__ __forceinline__ v8f wmma4(v2f a, v2f b, v8f c) {
  return __builtin_amdgcn_wmma_f32_16x16x4_f32(
      /*neg_a=*/false, a, /*neg_b=*/false, b,
      /*c_mod=*/(short)0, c, /*reuse_a=*/false, /*reuse_b=*/false);
}

// CDNA5 async global->LDS path (ASYNCcnt-tracked, bypasses VGPRs).
// Guarded: falls back to load/ds_store staging if the toolchain lacks the builtins.
#if defined(__has_builtin)
#if __has_builtin(__builtin_amdgcn_global_load_async_to_lds_b128) && \
    __has_builtin(__builtin_amdgcn_s_wait_asynccnt)
#define USE_ASYNC_LDS 1
#endif
#endif

#define AS1 __attribute__((address_space(1)))
#define AS3 __attribute__((address_space(3)))

#ifdef USE_ASYNC_LDS
// global (generic float*) -> LDS (generic float*) 16-byte async copy
__device__ __forceinline__ void async_cp16(const float* g, float* l) {
  __builtin_amdgcn_global_load_async_to_lds_b128(
      (AS1 v4i*)g, (AS3 v4i*)l, /*offset=*/0, /*cpol=*/0);
}
#endif

// ---------------------------------------------------------------------------
// Generic batched GEMM: C = A @ B (+ bias), fp32, WMMA 16x16x4.
// A[M,K] (lda), B[K,N] (ldb), C[M,N] (ldc). Requires M%64==0, N%64==0, K%32==0.
// Workgroup = 128 threads (4 waves, 2x2), tile 64x64, K staged 32 at a time.
// Async path: double-buffered LDS, GLOBAL_LOAD_ASYNC_TO_LDS_B128 overlapped
// with WMMA compute; one s_wait_asynccnt + one barrier per K-chunk.
// ---------------------------------------------------------------------------
#define TM 64
#define TN 64
#define TK 32

__global__ __launch_bounds__(128)
void gemm_f32_wmma(const float* __restrict__ A, const float* __restrict__ B,
                   const float* __restrict__ bias, float* __restrict__ C,
                   int M, int N, int K, int lda, int ldb, int ldc,
                   long sA, long sB, long sC)
{
#ifdef USE_ASYNC_LDS
  __shared__ float lsA[2][TM][TK + 4];   // pad 36: bank-spread + 16B-aligned rows
  __shared__ float lsB[2][TK][TN + 4];   // pad 68
#else
  __shared__ float lsA[1][TM][TK + 4];
  __shared__ float lsB[1][TK][TN + 4];
#endif
  const int bz   = blockIdx.z;
  A += (long)bz * sA;  B += (long)bz * sB;  C += (long)bz * sC;
  const int m0   = blockIdx.y * TM;
  const int n0   = blockIdx.x * TN;
  const int tid  = threadIdx.x;
  const int lane = tid & 31;
  const int wave = tid >> 5;
  const int wm   = wave >> 1, wn = wave & 1;
  const int l16  = lane & 15;
  const int kg   = (lane >> 4) << 1;     // K sub-offset: lanes 0-15 -> K{0,1}, 16-31 -> K{2,3}

  // This thread's four A-tile / four B-tile 16B chunks (same every K-chunk)
  const int arows[4] = { (tid + 0) >> 3, (tid + 128) >> 3, (tid + 256) >> 3, (tid + 384) >> 3 };
  const int acol     = (tid & 7) << 2;
  const int brows[4] = { (tid + 0) >> 4, (tid + 128) >> 4, (tid + 256) >> 4, (tid + 384) >> 4 };
  const int bcol     = (tid & 15) << 2;

  v8f acc[2][2] = {};

#ifdef USE_ASYNC_LDS
  // ---- async double-buffered pipeline ----
  {
    // prime buffer 0 with the first K-chunk
#pragma unroll
    for (int i = 0; i < 4; ++i) {
      async_cp16(A + (long)(m0 + arows[i]) * lda + acol, &lsA[0][arows[i]][acol]);
      async_cp16(B + (long)brows[i] * ldb + (n0 + bcol), &lsB[0][brows[i]][bcol]);
    }
    int cur = 0;
    for (int k0 = 0; k0 < K; k0 += TK) {
      __builtin_amdgcn_s_wait_asynccnt(0);   // my copies into lsX[cur] landed
      __syncthreads();                       // everyone's copies landed; prev compute done
      if (k0 + TK < K) {                     // prefetch next chunk into the other buffer
        const int nxt = cur ^ 1;
        const int kn  = k0 + TK;
#pragma unroll
        for (int i = 0; i < 4; ++i) {
          async_cp16(A + (long)(m0 + arows[i]) * lda + (kn + acol), &lsA[nxt][arows[i]][acol]);
          async_cp16(B + (long)(kn + brows[i]) * ldb + (n0 + bcol), &lsB[nxt][brows[i]][bcol]);
        }
      }
#pragma unroll
      for (int kk = 0; kk < TK; kk += 4) {
        int kf = kk + kg;
        v2f aF[2], bF[2];
#pragma unroll
        for (int mt = 0; mt < 2; ++mt) {
          int r = wm * 32 + mt * 16 + l16;
          aF[mt] = *reinterpret_cast<const v2f*>(&lsA[cur][r][kf]);
        }
#pragma unroll
        for (int nt = 0; nt < 2; ++nt) {
          int c = wn * 32 + nt * 16 + l16;
          v2f t;
          t.x = lsB[cur][kf][c];
          t.y = lsB[cur][kf + 1][c];
          bF[nt] = t;
        }
#pragma unroll
        for (int mt = 0; mt < 2; ++mt)
#pragma unroll
          for (int nt = 0; nt < 2; ++nt)
            acc[mt][nt] = wmma4(aF[mt], bF[nt], acc[mt][nt]);
      }
      cur ^= 1;
    }
  }
#else
  // ---- fallback: VGPR-staged tiles ----
  for (int k0 = 0; k0 < K; k0 += TK) {
#pragma unroll
    for (int i = 0; i < 4; ++i) {
      float4 val = *reinterpret_cast<const float4*>(A + (long)(m0 + arows[i]) * lda + (k0 + acol));
      *reinterpret_cast<float4*>(&lsA[0][arows[i]][acol]) = val;
    }
#pragma unroll
    for (int i = 0; i < 4; ++i) {
      float4 val = *reinterpret_cast<const float4*>(B + (long)(k0 + brows[i]) * ldb + (n0 + bcol));
      *reinterpret_cast<float4*>(&lsB[0][brows[i]][bcol]) = val;
    }
    __syncthreads();
#pragma unroll
    for (int kk = 0; kk < TK; kk += 4) {
      int kf = kk + kg;
      v2f aF[2], bF[2];
#pragma unroll
      for (int mt = 0; mt < 2; ++mt) {
        int r = wm * 32 + mt * 16 + l16;
        aF[mt] = *reinterpret_cast<const v2f*>(&lsA[0][r][kf]);
      }
#pragma unroll
      for (int nt = 0; nt < 2; ++nt) {
        int c = wn * 32 + nt * 16 + l16;
        v2f t;
        t.x = lsB[0][kf][c];
        t.y = lsB[0][kf + 1][c];
        bF[nt] = t;
      }
#pragma unroll
      for (int mt = 0; mt < 2; ++mt)
#pragma unroll
        for (int nt = 0; nt < 2; ++nt)
          acc[mt][nt] = wmma4(aF[mt], bF[nt], acc[mt][nt]);
    }
    __syncthreads();
  }
#endif

  // C/D layout: VGPR j -> row j (lanes 0-15) / j+8 (lanes 16-31); col = lane%16
#pragma unroll
  for (int mt = 0; mt < 2; ++mt) {
#pragma unroll
    for (int nt = 0; nt < 2; ++nt) {
#pragma unroll
      for (int j = 0; j < 8; ++j) {
        int row = m0 + wm * 32 + mt * 16 + j + ((lane >> 4) << 3);
        int col = n0 + wn * 32 + nt * 16 + l16;
        float vv = acc[mt][nt][j];
        if (bias) vv += bias[col];
        C[(long)row * ldc + col] = vv;
      }
    }
  }
}

// ---------------------------------------------------------------------------
// Fused multi-head attention adjacency: for one (batch, 16-query-row tile),
// loop all heads: scores = q.k^T * scale (WMMA), mask, row-softmax in LDS,
// accumulate head-sum, then store a[b, q0:q0+16, :].
// 128 threads: wave w owns key columns [64w, 64w+64).
// ---------------------------------------------------------------------------
__global__ __launch_bounds__(128)
void attn_softmax_sum(const float* __restrict__ q, const float* __restrict__ k,
                      const float* __restrict__ padf, float* __restrict__ aout,
                      int din, int dk, float scale)
{
  __shared__ float sc[16][NS + 1];
  __shared__ float accum[16][NS + 1];
  __shared__ float red[16][8];
  __shared__ float padq[16];
  __shared__ float padk[NS];

  const int b    = blockIdx.y;
  const int q0   = blockIdx.x * 16;
  const int tid  = threadIdx.x;
  const int lane = tid & 31;
  const int wave = tid >> 5;
  const int l16  = lane & 15;
  const int kg   = (lane >> 4) << 1;

  for (int i = tid; i < 16 * (NS + 1); i += 128) (&accum[0][0])[i] = 0.f;
  if (tid < 16) padq[tid] = padf[b * NS + q0 + tid];
  for (int c = tid; c < NS; c += 128) padk[c] = padf[b * NS + c];
  __syncthreads();

  const long qbase = ((long)b * NS + q0) * din;
  const long kbase = (long)b * NS * din;

  for (int h = 0; h < NHEADS; ++h) {
    const int hb = h * dk;
    v8f acc[4] = {};
    for (int kk = 0; kk < dk; kk += 4) {
      int kf = hb + kk + kg;
      v2f aF = *reinterpret_cast<const v2f*>(q + qbase + (long)l16 * din + kf);
#pragma unroll
      for (int nt = 0; nt < 4; ++nt) {
        int col = wave * 64 + nt * 16 + l16;
        // B(k x n) = K^T, so B[kf][col] = K[col][kf] -> contiguous v2f in K row
        v2f bF = *reinterpret_cast<const v2f*>(k + kbase + (long)col * din + kf);
        acc[nt] = wmma4(aF, bF, acc[nt]);
      }
    }
    // scaled + masked scores -> LDS
#pragma unroll
    for (int nt = 0; nt < 4; ++nt) {
#pragma unroll
      for (int j = 0; j < 8; ++j) {
        int row = j + ((lane >> 4) << 3);
        int col = wave * 64 + nt * 16 + l16;
        float s = acc[nt][j] * scale;
        if (padq[row] > 0.5f || padk[col] > 0.5f) s = -1e9f;
        sc[row][col] = s;
      }
    }
    __syncthreads();

    // row softmax: 8 threads per row, 32 cols each
    const int rr = tid >> 3;
    const int tg = tid & 7;
    float mx = -3.4e38f;
    for (int c = tg * 32; c < tg * 32 + 32; ++c) mx = fmaxf(mx, sc[rr][c]);
    red[rr][tg] = mx;
    __syncthreads();
    float fmx = -3.4e38f;
#pragma unroll
    for (int t = 0; t < 8; ++t) fmx = fmaxf(fmx, red[rr][t]);
    __syncthreads();
    float psum = 0.f;
    for (int c = tg * 32; c < tg * 32 + 32; ++c) {
      float e = __expf(sc[rr][c] - fmx);
      sc[rr][c] = e;
      psum += e;
    }
    red[rr][tg] = psum;
    __syncthreads();
    float tot = 0.f;
#pragma unroll
    for (int t = 0; t < 8; ++t) tot += red[rr][t];
    float inv = (padq[rr] > 0.5f) ? 0.f : (1.f / tot);   // keep-mask zeroes pad rows
    for (int c = tg * 32; c < tg * 32 + 32; ++c) accum[rr][c] += sc[rr][c] * inv;
    __syncthreads();
  }

  for (int i = tid; i < 16 * NS; i += 128) {
    int row = i >> 8, col = i & 255;
    aout[((long)b * NS + q0 + row) * NS + col] = accum[row][col];
  }
}

// ---------------------------------------------------------------------------
// Exact per-batch top-k threshold via 4-pass radix select over uint bit
// patterns (valid: all scores are non-negative floats).
// ---------------------------------------------------------------------------
__global__ __launch_bounds__(256)
void topk_threshold(const float* __restrict__ a, float* __restrict__ thr)
{
  __shared__ unsigned int hist[256];
  __shared__ unsigned int sbin, srem;
  const int b   = blockIdx.x;
  const int tid = threadIdx.x;
  const unsigned int* av = reinterpret_cast<const unsigned int*>(a) + (long)b * NS * NS;
  unsigned int prefix = 0, pmask = 0;
  unsigned int rem = NTOPK;
  for (int shift = 24; shift >= 0; shift -= 8) {
    hist[tid] = 0;
    __syncthreads();
    for (int i = tid; i < NS * NS; i += 256) {
      unsigned int v = av[i];
      if ((v & pmask) == prefix) atomicAdd(&hist[(v >> shift) & 0xFF], 1u);
    }
    __syncthreads();
    if (tid == 0) {
      unsigned int cum = 0; unsigned int bin = 0; unsigned int r = rem;
      for (int bb = 255; bb >= 0; --bb) {
        unsigned int c = hist[bb];
        if (cum + c >= r) { bin = (unsigned)bb; r -= cum; break; }
        cum += c;
      }
      sbin = bin; srem = r;
    }
    __syncthreads();
    prefix |= sbin << shift;
    pmask  |= 0xFFu << shift;
    rem     = srem;
    __syncthreads();
  }
  if (tid == 0) thr[b] = __uint_as_float(prefix);
}

// newA[i,j] = (i==j) ? a[i,j] : ((a[i,j]>=thr) + (a[j,i]>=thr)) * a[i,j]
// (reference's sel + sel^T is NOT clipped; off-diag coefficient may be 2)
__global__ void select_apply(const float* __restrict__ a, const float* __restrict__ thr,
                             float* __restrict__ nA)
{
  long idx = (long)blockIdx.x * 256 + threadIdx.x;
  int b = (int)(idx >> 16);
  int rest = (int)(idx & 65535);
  int i = rest >> 8, j = rest & 255;
  const float* ab = a + ((long)b << 16);
  float t   = thr[b];
  float aij = ab[i * NS + j];
  float outv;
  if (i == j) {
    outv = aij;
  } else {
    float aji = ab[j * NS + i];
    float s = (aij >= t ? 1.f : 0.f) + (aji >= t ? 1.f : 0.f);
    outv = s * aij;
  }
  nA[idx] = outv;
}

// denom[row] = sum_j A[row, j] + 1 ; one wave per row
__global__ __launch_bounds__(256)
void rowsum_kernel(const float* __restrict__ A, float* __restrict__ denom, int ncols)
{
  int row  = blockIdx.x * 8 + (threadIdx.x >> 5);
  int lane = threadIdx.x & 31;
  const float* r = A + (long)row * ncols;
  float s = 0.f;
  for (int c = lane; c < ncols; c += 32) s += r[c];
#pragma unroll
  for (int m = 16; m; m >>= 1) s += __shfl_xor(s, m, 32);
  if (lane == 0) denom[row] = s + 1.0f;
}

// out = relu(u/denom) + v   (u, v already have bias folded in)
__global__ void combine_kernel(const float* __restrict__ u, const float* __restrict__ v,
                               const float* __restrict__ denom,
                               float* __restrict__ out, int ldout)
{
  long s = blockIdx.x;
  int  m = threadIdx.x;
  long idx = s * NMEM + m;
  float d = denom[s];
  out[s * (long)ldout + m] = fmaxf(u[idx] / d, 0.f) + v[idx];
}

__global__ void copy_in_kernel(const float* __restrict__ in, float* __restrict__ xbuf)
{
  long idx = (long)blockIdx.x * 256 + threadIdx.x;   // over B*S*512
  long s = idx >> 9;
  int  c = (int)(idx & 511);
  xbuf[s * 1024 + c] = in[idx];
}

// pad[b,s] = score_mask[b,0,s,s]  (diag of pad|pad == pad)
__global__ void pad_extract_kernel(const unsigned char* __restrict__ m, float* __restrict__ padf)
{
  int idx = blockIdx.x * 256 + threadIdx.x;          // over B*S
  int b = idx >> 8, s = idx & 255;
  padf[idx] = m[(long)b * NS * NS + (long)s * (NS + 1)] ? 1.f : 0.f;
}

// ---------------------------------------------------------------------------
static inline void launch_gemm(const float* A, const float* B, const float* bias, float* C,
                               int M, int N, int K, int lda, int ldb, int ldc,
                               long sA, long sB, long sC, int batch, hipStream_t st)
{
  dim3 g((unsigned)(N / TN), (unsigned)(M / TM), (unsigned)batch);
  gemm_f32_wmma<<<g, dim3(128), 0, st>>>(A, B, bias, C, M, N, K, lda, ldb, ldc, sA, sB, sC);
}

extern "C" void kernel_launch(void* const* d_in, const int* in_sizes, int n_in,
                              void* d_out, int out_size, void* d_ws, size_t ws_size,
                              hipStream_t stream)
{
  (void)in_sizes; (void)n_in; (void)out_size; (void)ws_size;

  const float* adj    = (const float*)d_in[0];
  const float* inputs = (const float*)d_in[1];
  const unsigned char* smask = (const unsigned char*)d_in[2];
  const float* W_w[3] = {(const float*)d_in[3], (const float*)d_in[5], (const float*)d_in[7]};
  const float* W_b[3] = {(const float*)d_in[4], (const float*)d_in[6], (const float*)d_in[8]};
  const float* qw[2]  = {(const float*)d_in[9],  (const float*)d_in[13]};
  const float* qb[2]  = {(const float*)d_in[10], (const float*)d_in[14]};
  const float* kw[2]  = {(const float*)d_in[11], (const float*)d_in[15]};
  const float* kb[2]  = {(const float*)d_in[12], (const float*)d_in[16]};

  const long BS = (long)NB * NS;
  float* ws    = (float*)d_ws;
  float* xbuf  = ws;                       // [BS,1024] growing concat features
  float* qbuf  = xbuf + BS * 1024;         // [BS,1024] q / t scratch
  float* kbuf  = qbuf + BS * 1024;         // [BS,1024] k / (u,v) scratch
  float* abuf  = kbuf + BS * 1024;         // [B,S,S] attention adjacency
  float* nAbuf = abuf + (long)NB * NS * NS;
  float* denom = nAbuf + (long)NB * NS * NS; // [BS]
  float* padfb = denom + BS;               // [BS]
  float* thr   = padfb + BS;               // [B]
  float* ubuf  = kbuf;                     // aliases (k dead when used)
  float* vbuf  = kbuf + BS * NMEM;

  copy_in_kernel<<<(unsigned)(BS * 512 / 256), 256, 0, stream>>>(inputs, xbuf);
  pad_extract_kernel<<<(unsigned)(BS / 256), 256, 0, stream>>>(smask, padfb);

  // ----- layer 0: out0 = relu((adj@x)@W0+b0)/denom + (x@W0+b0), x = inputs -----
  rowsum_kernel<<<(unsigned)(BS / 8), 256, 0, stream>>>(adj, denom, NS);
  // t0 = adj @ x512 (batched per b)
  launch_gemm(adj, xbuf, nullptr, qbuf, NS, NIN, NS, NS, 1024, 1024,
              (long)NS * NS, (long)NS * 1024, (long)NS * 1024, NB, stream);
  launch_gemm(qbuf, W_w[0], W_b[0], ubuf, (int)BS, NMEM, NIN, 1024, NMEM, NMEM, 0, 0, 0, 1, stream);
  launch_gemm(xbuf, W_w[0], W_b[0], vbuf, (int)BS, NMEM, NIN, 1024, NMEM, NMEM, 0, 0, 0, 1, stream);
  combine_kernel<<<(unsigned)BS, NMEM, 0, stream>>>(ubuf, vbuf, denom, xbuf + NIN, 1024);

  // ----- layers 1,2 with attention rewiring -----
  const int dins[2] = {NIN + NMEM, NIN + 2 * NMEM};   // 768, 1024
  for (int li = 0; li < 2; ++li) {
    const int din = dins[li];
    const int dk  = din / NHEADS;                      // 96, 128
    const float scale = 1.0f / sqrtf((float)dk);

    launch_gemm(xbuf, qw[li], qb[li], qbuf, (int)BS, din, din, 1024, din, din, 0, 0, 0, 1, stream);
    launch_gemm(xbuf, kw[li], kb[li], kbuf, (int)BS, din, din, 1024, din, din, 0, 0, 0, 1, stream);

    attn_softmax_sum<<<dim3(NS / 16, NB), 128, 0, stream>>>(qbuf, kbuf, padfb, abuf, din, dk, scale);
    topk_threshold<<<NB, 256, 0, stream>>>(abuf, thr);
    select_apply<<<(unsigned)((long)NB * NS * NS / 256), 256, 0, stream>>>(abuf, thr, nAbuf);
    rowsum_kernel<<<(unsigned)(BS / 8), 256, 0, stream>>>(nAbuf, denom, NS);

    // t = newA @ x (batched), then u = t@W+b, v = x@W+b
    launch_gemm(nAbuf, xbuf, nullptr, qbuf, NS, din, NS, NS, 1024, 1024,
                (long)NS * NS, (long)NS * 1024, (long)NS * 1024, NB, stream);
    launch_gemm(qbuf, W_w[li + 1], W_b[li + 1], ubuf, (int)BS, NMEM, din, 1024, NMEM, NMEM, 0, 0, 0, 1, stream);
    launch_gemm(xbuf, W_w[li + 1], W_b[li + 1], vbuf, (int)BS, NMEM, din, 1024, NMEM, NMEM, 0, 0, 0, 1, stream);

    float* outp = (li == 0) ? (xbuf + NIN + NMEM) : (float*)d_out;
    int    ldo  = (li == 0) ? 1024 : NMEM;
    combine_kernel<<<(unsigned)BS, NMEM, 0, stream>>>(ubuf, vbuf, denom, outp, ldo);
  }
}